// GATConv2d_21328807592398
// MI455X (gfx1250) — compile-verified
//
#include <hip/hip_runtime.h>
#include <hip/hip_bf16.h>
#include <cstdint>
#include <cstddef>

// Problem constants (match reference)
#define B_      4
#define C_IN_   128
#define C_OUT_  128
#define N_      10000
#define K_      16
#define KP1_    17          // K neighbors + self edge
#define NEG_SLOPE_ 0.2f

#define XS_STRIDE 132       // floats per LDS row; 132 % 64 == 4 -> conflict-free b64 reads

typedef __attribute__((ext_vector_type(2))) float v2f;
typedef __attribute__((ext_vector_type(8))) float v8f;

// ---------------------------------------------------------------------------
// Kernel A: h_t[b][n][o] = relu( sum_c W[o][c] * x[b][c][n] )
// Stored node-major (o fastest) so later gathers are contiguous 512B reads.
// One block = 128 o x 32 n tile; 8 waves, each wave owns a 16-o slice and
// two 16-n subtiles; K-loop of 32 V_WMMA_F32_16X16X4_F32 steps per subtile.
// LDS x tile is stored transposed [n][c] so each B fragment is ONE ds_load_b64.
// ---------------------------------------------------------------------------
__global__ __launch_bounds__(256) void gat_gemm_relu(
    const float* __restrict__ x,      // (B, C_IN, N)
    const float* __restrict__ W,      // (C_OUT, C_IN) row-major
    float* __restrict__ h_t)          // (B, N, C_OUT)
{
    __shared__ float xs[32 * XS_STRIDE];   // 32 n-rows x 128 c (padded), ~16.9KB

    const int n_base = blockIdx.x * 32;
    const int b      = blockIdx.y;
    const int t      = threadIdx.x;

    // Cooperative stage: global read coalesced in n, transposed write to LDS.
    for (int idx = t; idx < C_IN_ * 32; idx += 256) {
        const int c = idx >> 5;
        const int j = idx & 31;
        int n = n_base + j;
        if (n >= N_) n = N_ - 1;      // clamp: garbage columns never stored
        xs[j * XS_STRIDE + c] = x[(size_t)b * C_IN_ * N_ + (size_t)c * N_ + n];
    }
    __syncthreads();

    const int wave   = t >> 5;
    const int lane   = t & 31;
    const int o_base = wave * 16;
    const int m      = lane & 15;          // A: M row; B: N column
    const int hi     = lane >> 4;
    const int k0     = hi << 1;            // 0 (lanes 0-15) or 2 (lanes 16-31)

    v8f acc0 = {0.f,0.f,0.f,0.f,0.f,0.f,0.f,0.f};
    v8f acc1 = {0.f,0.f,0.f,0.f,0.f,0.f,0.f,0.f};

    const float* wrow = W + (size_t)(o_base + m) * C_IN_;
    const float* xr0  = xs + (size_t)m * XS_STRIDE;          // n = n_base + m
    const float* xr1  = xr0 + 16 * XS_STRIDE;                // n = n_base + 16 + m

    #pragma unroll 4
    for (int c = 0; c < C_IN_; c += 4) {
        // A fragment 16x4 (M=o, K=c): VGPR0 = K{0|2}, VGPR1 = K{1|3}
        const v2f a  = *reinterpret_cast<const v2f*>(wrow + c + k0);
        // B fragments 4x16: contiguous (k, k+1) pair -> single ds_load_b64
        const v2f b0 = *reinterpret_cast<const v2f*>(xr0 + c + k0);
        const v2f b1 = *reinterpret_cast<const v2f*>(xr1 + c + k0);
        acc0 = __builtin_amdgcn_wmma_f32_16x16x4_f32(false, a, false, b0, (short)0, acc0, false, false);
        acc1 = __builtin_amdgcn_wmma_f32_16x16x4_f32(false, a, false, b1, (short)0, acc1, false, false);
    }

    // C/D layout: VGPR v, lanes 0-15 -> M=v; lanes 16-31 -> M=v+8 (col = m).
    // For one lane the 8 acc elements are 8 CONSECUTIVE o values -> 2 b128 stores.
    const int n0 = n_base + m;
    const int n1 = n_base + 16 + m;
    const int o0 = o_base + 8 * hi;

    float4 r0lo = make_float4(fmaxf(acc0[0],0.f), fmaxf(acc0[1],0.f), fmaxf(acc0[2],0.f), fmaxf(acc0[3],0.f));
    float4 r0hi = make_float4(fmaxf(acc0[4],0.f), fmaxf(acc0[5],0.f), fmaxf(acc0[6],0.f), fmaxf(acc0[7],0.f));
    float4 r1lo = make_float4(fmaxf(acc1[0],0.f), fmaxf(acc1[1],0.f), fmaxf(acc1[2],0.f), fmaxf(acc1[3],0.f));
    float4 r1hi = make_float4(fmaxf(acc1[4],0.f), fmaxf(acc1[5],0.f), fmaxf(acc1[6],0.f), fmaxf(acc1[7],0.f));

    float* p0 = h_t + ((size_t)b * N_ + n0) * C_OUT_ + o0;
    float* p1 = h_t + ((size_t)b * N_ + n1) * C_OUT_ + o0;

    if (n_base + 32 <= N_) {               // uniform fast path (312 of 313 blocks)
        *reinterpret_cast<float4*>(p0)     = r0lo;
        *reinterpret_cast<float4*>(p0 + 4) = r0hi;
        *reinterpret_cast<float4*>(p1)     = r1lo;
        *reinterpret_cast<float4*>(p1 + 4) = r1hi;
    } else {                               // edge block: per-lane guards
        if (n0 < N_) {
            *reinterpret_cast<float4*>(p0)     = r0lo;
            *reinterpret_cast<float4*>(p0 + 4) = r0hi;
        }
        if (n1 < N_) {
            *reinterpret_cast<float4*>(p1)     = r1lo;
            *reinterpret_cast<float4*>(p1 + 4) = r1hi;
        }
    }
}

// ---------------------------------------------------------------------------
// Kernel B: per-node attention scalars
//   a_i[b*N+n] = att[0:128]   . h_t[b][n][:]
//   a_j[b*N+n] = att[128:256] . h_t[b][n][:]
// ---------------------------------------------------------------------------
__global__ __launch_bounds__(256) void gat_att_dots(
    const float* __restrict__ h_t,
    const float* __restrict__ att,
    float* __restrict__ a_i,
    float* __restrict__ a_j)
{
    const int idx = blockIdx.x * 256 + threadIdx.x;   // idx = b*N + n
    if (idx >= B_ * N_) return;
    const float4* col  = reinterpret_cast<const float4*>(h_t + (size_t)idx * C_OUT_);
    const float4* atti = reinterpret_cast<const float4*>(att);
    const float4* attj = reinterpret_cast<const float4*>(att + C_OUT_);
    float si = 0.f, sj = 0.f;
    #pragma unroll 8
    for (int q = 0; q < C_OUT_ / 4; ++q) {
        const float4 v  = col[q];
        const float4 ai = atti[q];
        const float4 aj = attj[q];
        si = fmaf(ai.x, v.x, fmaf(ai.y, v.y, fmaf(ai.z, v.z, fmaf(ai.w, v.w, si))));
        sj = fmaf(aj.x, v.x, fmaf(aj.y, v.y, fmaf(aj.z, v.z, fmaf(aj.w, v.w, sj))));
    }
    a_i[idx] = si;
    a_j[idx] = sj;
}

// ---------------------------------------------------------------------------
// Kernel C: one wave32 per node.
//  - All lanes redundantly compute the 17 logits (broadcast index loads),
//    leaky-relu, numerically-stable softmax held in registers.
//  - Each lane owns 4 channels; each neighbor gather is one fully-coalesced
//    512B float4 wave read from node-major h_t (L2-resident, 192MB L2).
// ---------------------------------------------------------------------------
__global__ __launch_bounds__(256) void gat_aggregate(
    const float* __restrict__ h_t,
    const float* __restrict__ a_i,
    const float* __restrict__ a_j,
    const long long* __restrict__ eidx,   // (2, B, N, K) int64
    const float* __restrict__ bias,
    float* __restrict__ out)              // (B, C_OUT, N)
{
    const int lane = threadIdx.x & 31;
    const int node = blockIdx.x * 8 + (threadIdx.x >> 5);  // wave-uniform
    if (node >= B_ * N_) return;
    const int b = node / N_;
    const int n = node - b * N_;

    const long long* ej_p = eidx + ((size_t)b        * N_ + n) * K_;  // source (j)
    const long long* ei_p = eidx + ((size_t)(B_ + b) * N_ + n) * K_;  // target (i)

    int   jj[KP1_];
    float lg[KP1_];
    float mx = -1e30f;
    #pragma unroll
    for (int k = 0; k < KP1_; ++k) {
        const int j = (k < K_) ? (int)ej_p[k] : n;   // self edge at k=16
        const int i = (k < K_) ? (int)ei_p[k] : n;
        jj[k] = j;
        float l = a_i[b * N_ + i] + a_j[b * N_ + j];
        l = (l >= 0.f) ? l : NEG_SLOPE_ * l;
        lg[k] = l;
        mx = fmaxf(mx, l);
    }
    float s = 0.f;
    #pragma unroll
    for (int k = 0; k < KP1_; ++k) {
        lg[k] = __expf(lg[k] - mx);
        s += lg[k];
    }
    const float inv = 1.f / s;

    float4 acc = make_float4(0.f, 0.f, 0.f, 0.f);
    #pragma unroll
    for (int k = 0; k < KP1_; ++k) {
        const float4 hv = *reinterpret_cast<const float4*>(
            h_t + ((size_t)b * N_ + jj[k]) * C_OUT_ + 4 * lane);
        const float w = lg[k] * inv;
        acc.x = fmaf(w, hv.x, acc.x);
        acc.y = fmaf(w, hv.y, acc.y);
        acc.z = fmaf(w, hv.z, acc.z);
        acc.w = fmaf(w, hv.w, acc.w);
    }

    const int c0 = 4 * lane;
    const size_t ob = (size_t)b * C_OUT_ * N_;
    out[ob + (size_t)(c0 + 0) * N_ + n] = acc.x + bias[c0 + 0];
    out[ob + (size_t)(c0 + 1) * N_ + n] = acc.y + bias[c0 + 1];
    out[ob + (size_t)(c0 + 2) * N_ + n] = acc.z + bias[c0 + 2];
    out[ob + (size_t)(c0 + 3) * N_ + n] = acc.w + bias[c0 + 3];
}

// ---------------------------------------------------------------------------
extern "C" void kernel_launch(void* const* d_in, const int* in_sizes, int n_in,
                              void* d_out, int out_size, void* d_ws, size_t ws_size,
                              hipStream_t stream) {
    const float*     x      = (const float*)d_in[0];      // (B, C_IN, N, 1) f32
    const long long* eidx   = (const long long*)d_in[1];  // (2, B, N, K) i64
    const float*     conv_w = (const float*)d_in[2];      // (C_OUT, C_IN) f32
    const float*     att    = (const float*)d_in[3];      // (2*C_OUT,) f32
    const float*     bias   = (const float*)d_in[4];      // (C_OUT,) f32
    float*           out    = (float*)d_out;              // (B, C_OUT, N, 1) f32

    // Workspace layout: h_t (B*N*C_OUT f32, 20.5MB) | a_i (B*N) | a_j (B*N)
    float* h_t = (float*)d_ws;
    float* a_i = h_t + (size_t)B_ * N_ * C_OUT_;
    float* a_j = a_i + (size_t)B_ * N_;

    dim3 gA((N_ + 31) / 32, B_);
    gat_gemm_relu<<<gA, 256, 0, stream>>>(x, conv_w, h_t);

    gat_att_dots<<<(B_ * N_ + 255) / 256, 256, 0, stream>>>(h_t, att, a_i, a_j);

    gat_aggregate<<<(B_ * N_ + 7) / 8, 256, 0, stream>>>(h_t, a_i, a_j, eidx, bias, out);
}